// GATNet_5566277616456
// MI455X (gfx1250) — compile-verified
//
#include <hip/hip_runtime.h>
#include <math.h>

// ---------------------------------------------------------------------------
// GATNet on MI455X (gfx1250): fp32 WMMA GEMMs (V_WMMA_F32_16X16X4_F32) with a
// fused attention-score epilogue, plus atomic-based segment softmax/aggregate.
// Wave32 throughout. Each GEMM wave computes a 32x64 output tile (8 WMMA
// accumulators) so every B fragment load feeds two WMMAs.
// ---------------------------------------------------------------------------

typedef __attribute__((ext_vector_type(2))) float v2f;
typedef __attribute__((ext_vector_type(8))) float v8f;

#define HEADS     4
#define CHAN      64
#define HIDDEN    256          // HEADS*CHAN
#define FEAT      128
#define NEG_SLOPE 0.2f
#define ORD_NEG_INF 0x007FFFFFu   // ordEnc(-inf)

static __device__ __forceinline__ float lrelu(float x) {
  return x > 0.f ? x : NEG_SLOPE * x;
}
// Order-preserving float<->uint encoding so atomicMax(u32) == float max.
static __device__ __forceinline__ unsigned ordEnc(float f) {
  unsigned u = __float_as_uint(f);
  return (u & 0x80000000u) ? ~u : (u | 0x80000000u);
}
static __device__ __forceinline__ float ordDec(unsigned u) {
  unsigned v = (u & 0x80000000u) ? (u & 0x7fffffffu) : ~u;
  return __uint_as_float(v);
}

// ---------------------------------------------------------------------------
// GEMM: Hout[M,Nc] = A[M,K] @ W[K,Nc] (+bias).  One wave -> 32x64 tile.
// GAT epilogue: per-head score dots s_src/s_dst via in-tile cross-lane reduce
// (a 64-col tile == exactly one head since CHAN == 64).
// Requires M % 32 == 0, K % 4 == 0, Nc % 64 == 0 (true here: 20000/128/256/64).
// ---------------------------------------------------------------------------
template <bool GAT, bool BIAS>
__global__ void wmma_gemm_kernel(const float* __restrict__ A,
                                 const float* __restrict__ W,
                                 const float* __restrict__ bias,
                                 const float* __restrict__ att_src,
                                 const float* __restrict__ att_dst,
                                 float* __restrict__ Hout,
                                 float* __restrict__ s_src,
                                 float* __restrict__ s_dst,
                                 int M, int K, int Nc) {
  const int lane = threadIdx.x & 31;
  const int wv   = threadIdx.x >> 5;
  const int wpb  = blockDim.x >> 5;
  const int tile = blockIdx.x * wpb + wv;
  const int ctiles = Nc >> 6;
  const int mt = tile / ctiles;
  const int ct = tile - mt * ctiles;
  const int m0 = mt << 5;                // 32 rows per wave tile
  if (m0 >= M) return;
  const int n0 = ct << 6;
  const int lo = lane & 15;   // row (A) / col (B,C) within 16x16 sub-tile
  const int hi = lane >> 4;   // selects K pair (A,B) / row half (C)

  v8f acc[2][4] = {};                    // [row-group][col-group]
  const float* arow0 = A + (size_t)(m0 + lo) * K;        // rows m0..m0+15
  const float* arow1 = A + (size_t)(m0 + 16 + lo) * K;   // rows m0+16..m0+31
  const float* wcol  = W + n0 + lo;

  for (int k = 0; k < K; k += 4) {
    const int ka = k + (hi << 1);
    v2f a0 = *(const v2f*)(arow0 + ka);        // A frags: K=ka..ka+1
    v2f a1 = *(const v2f*)(arow1 + ka);
    const float* w0 = wcol + (size_t)ka * Nc;  // B frag rows K=ka, ka+1
    const float* w1 = w0 + Nc;
    v2f b0 = {w0[0],  w1[0]};
    v2f b1 = {w0[16], w1[16]};
    v2f b2 = {w0[32], w1[32]};
    v2f b3 = {w0[48], w1[48]};
    acc[0][0] = __builtin_amdgcn_wmma_f32_16x16x4_f32(false, a0, false, b0, (short)0, acc[0][0], false, false);
    acc[1][0] = __builtin_amdgcn_wmma_f32_16x16x4_f32(false, a1, false, b0, (short)0, acc[1][0], false, false);
    acc[0][1] = __builtin_amdgcn_wmma_f32_16x16x4_f32(false, a0, false, b1, (short)0, acc[0][1], false, false);
    acc[1][1] = __builtin_amdgcn_wmma_f32_16x16x4_f32(false, a1, false, b1, (short)0, acc[1][1], false, false);
    acc[0][2] = __builtin_amdgcn_wmma_f32_16x16x4_f32(false, a0, false, b2, (short)0, acc[0][2], false, false);
    acc[1][2] = __builtin_amdgcn_wmma_f32_16x16x4_f32(false, a1, false, b2, (short)0, acc[1][2], false, false);
    acc[0][3] = __builtin_amdgcn_wmma_f32_16x16x4_f32(false, a0, false, b3, (short)0, acc[0][3], false, false);
    acc[1][3] = __builtin_amdgcn_wmma_f32_16x16x4_f32(false, a1, false, b3, (short)0, acc[1][3], false, false);
  }

  float bv0 = 0.f, bv1 = 0.f, bv2 = 0.f, bv3 = 0.f;
  if (BIAS) {
    bv0 = bias[n0 + lo];      bv1 = bias[n0 + 16 + lo];
    bv2 = bias[n0 + 32 + lo]; bv3 = bias[n0 + 48 + lo];
  }

  #pragma unroll
  for (int g = 0; g < 2; ++g) {
    const int rbase = m0 + (g << 4) + (hi << 3);
    #pragma unroll
    for (int r = 0; r < 8; ++r) {
      float* hp = Hout + (size_t)(rbase + r) * Nc + n0 + lo;
      hp[0]  = acc[g][0][r] + bv0;
      hp[16] = acc[g][1][r] + bv1;
      hp[32] = acc[g][2][r] + bv2;
      hp[48] = acc[g][3][r] + bv3;
    }
  }

  if (GAT) {
    const int head = n0 >> 6;                 // 64-col tile == one head
    const float* as = att_src + head * CHAN;
    const float* ad = att_dst + head * CHAN;
    const float as0 = as[lo], as1 = as[16 + lo], as2 = as[32 + lo], as3 = as[48 + lo];
    const float ad0 = ad[lo], ad1 = ad[16 + lo], ad2 = ad[32 + lo], ad3 = ad[48 + lo];
    #pragma unroll
    for (int g = 0; g < 2; ++g) {
      const int rbase = m0 + (g << 4) + (hi << 3);
      #pragma unroll
      for (int r = 0; r < 8; ++r) {
        float ss = acc[g][0][r] * as0 + acc[g][1][r] * as1 + acc[g][2][r] * as2 + acc[g][3][r] * as3;
        float sd = acc[g][0][r] * ad0 + acc[g][1][r] * ad1 + acc[g][2][r] * ad2 + acc[g][3][r] * ad3;
        // reduce over the 16 columns held in this half-wave (xor masks stay in-group)
        #pragma unroll
        for (int off = 8; off > 0; off >>= 1) {
          ss += __shfl_xor(ss, off, 32);
          sd += __shfl_xor(sd, off, 32);
        }
        if (lo == 0) {
          const int row = rbase + r;
          s_src[row * HEADS + head] = ss;
          s_dst[row * HEADS + head] = sd;
        }
      }
    }
  }
}

// out accumulator <- bias; denom <- 0; running-max <- ord(-inf)
__global__ void layer_init_kernel(float* __restrict__ agg, const float* __restrict__ bias,
                                  float* __restrict__ denom, unsigned* __restrict__ mord,
                                  int N) {
  const int idx = blockIdx.x * blockDim.x + threadIdx.x;
  if (idx < N * HIDDEN) agg[idx] = bias[idx & (HIDDEN - 1)];
  if (idx < N * HEADS) { denom[idx] = 0.f; mord[idx] = ORD_NEG_INF; }
}

static __device__ __forceinline__ void edge_nodes(const int* __restrict__ ei,
                                                  int idx, int E, int& s, int& d) {
  if (idx < E) { s = ei[idx]; d = ei[E + idx]; }
  else         { s = d = idx - E; }            // self loops appended
}

// pass 1: segment max of leaky_relu(s_src[src]+s_dst[dst]) per (dst, head)
__global__ void edge_max_kernel(const int* __restrict__ ei, int E, int N,
                                const float* __restrict__ s_src,
                                const float* __restrict__ s_dst,
                                unsigned* __restrict__ mord) {
  const int idx = blockIdx.x * blockDim.x + threadIdx.x;
  if (idx >= E + N) return;
  int s, d; edge_nodes(ei, idx, E, s, d);
  const float4 vs = *(const float4*)(s_src + (size_t)s * HEADS);
  const float4 vd = *(const float4*)(s_dst + (size_t)d * HEADS);
  unsigned* mp = mord + (size_t)d * HEADS;
  atomicMax(mp + 0, ordEnc(lrelu(vs.x + vd.x)));
  atomicMax(mp + 1, ordEnc(lrelu(vs.y + vd.y)));
  atomicMax(mp + 2, ordEnc(lrelu(vs.z + vd.z)));
  atomicMax(mp + 3, ordEnc(lrelu(vs.w + vd.w)));
}

// pass 2: ex = exp(e - max), denom = segment sum
__global__ void edge_exp_kernel(const int* __restrict__ ei, int E, int N,
                                const float* __restrict__ s_src,
                                const float* __restrict__ s_dst,
                                const unsigned* __restrict__ mord,
                                float* __restrict__ exbuf,
                                float* __restrict__ denom) {
  const int idx = blockIdx.x * blockDim.x + threadIdx.x;
  if (idx >= E + N) return;
  int s, d; edge_nodes(ei, idx, E, s, d);
  const float4 vs = *(const float4*)(s_src + (size_t)s * HEADS);
  const float4 vd = *(const float4*)(s_dst + (size_t)d * HEADS);
  const unsigned* mp = mord + (size_t)d * HEADS;
  const float x0 = expf(lrelu(vs.x + vd.x) - ordDec(mp[0]));
  const float x1 = expf(lrelu(vs.y + vd.y) - ordDec(mp[1]));
  const float x2 = expf(lrelu(vs.z + vd.z) - ordDec(mp[2]));
  const float x3 = expf(lrelu(vs.w + vd.w) - ordDec(mp[3]));
  *(float4*)(exbuf + (size_t)idx * HEADS) = make_float4(x0, x1, x2, x3);
  float* dn = denom + (size_t)d * HEADS;
  atomicAdd(dn + 0, x0);
  atomicAdd(dn + 1, x1);
  atomicAdd(dn + 2, x2);
  atomicAdd(dn + 3, x3);
}

// pass 3: one wave per edge; scatter-add alpha * h[src] into out[dst]
__global__ void edge_agg_kernel(const int* __restrict__ ei, int E, int N,
                                const float* __restrict__ exbuf,
                                const float* __restrict__ denom,
                                const float* __restrict__ Hin,
                                float* __restrict__ agg) {
  const int wid  = blockIdx.x * (blockDim.x >> 5) + (threadIdx.x >> 5);
  const int lane = threadIdx.x & 31;
  if (wid >= E + N) return;
  int s, d; edge_nodes(ei, wid, E, s, d);
  const float4 e4 = *(const float4*)(exbuf + (size_t)wid * HEADS);
  const float4 d4 = *(const float4*)(denom + (size_t)d * HEADS);
  const float alpha[HEADS] = {e4.x / d4.x, e4.y / d4.y, e4.z / d4.z, e4.w / d4.w};
  const float* hs = Hin + (size_t)s * HIDDEN;
  float* od = agg + (size_t)d * HIDDEN;
  #pragma unroll
  for (int i = 0; i < 8; ++i) {
    const int ch = lane + (i << 5);            // head = ch>>6 = i>>1 (compile-time)
    atomicAdd(od + ch, hs[ch] * alpha[i >> 1]);
  }
}

__global__ void elu_kernel(float* __restrict__ x, int n) {
  const int idx = blockIdx.x * blockDim.x + threadIdx.x;
  if (idx < n) {
    const float v = x[idx];
    x[idx] = v > 0.f ? v : expm1f(v);
  }
}

// ---------------------------------------------------------------------------
extern "C" void kernel_launch(void* const* d_in, const int* in_sizes, int n_in,
                              void* d_out, int out_size, void* d_ws, size_t ws_size,
                              hipStream_t stream) {
  const float* x     = (const float*)d_in[0];
  const int*   ei    = (const int*)  d_in[1];
  const float* W1    = (const float*)d_in[2];
  const float* asrc1 = (const float*)d_in[3];
  const float* adst1 = (const float*)d_in[4];
  const float* b1    = (const float*)d_in[5];
  const float* W2    = (const float*)d_in[6];
  const float* asrc2 = (const float*)d_in[7];
  const float* adst2 = (const float*)d_in[8];
  const float* b2    = (const float*)d_in[9];
  const float* Wf    = (const float*)d_in[10];
  const float* bf    = (const float*)d_in[11];

  const int B  = 2;
  const int N  = in_sizes[0] / (B * FEAT);   // 20000
  const int E  = in_sizes[1] / 2;            // 320000
  const int ET = E + N;

  // workspace layout (all fp32; ~48 MB total)
  float* p = (float*)d_ws;
  float* buf_h   = p; p += (size_t)N * HIDDEN;   // per-layer h = X@W
  float* buf_agg = p; p += (size_t)N * HIDDEN;   // aggregated output / next input
  float* ssrc    = p; p += (size_t)N * HEADS;
  float* sdst    = p; p += (size_t)N * HEADS;
  float* denom   = p; p += (size_t)N * HEADS;
  unsigned* mord = (unsigned*)p; p += (size_t)N * HEADS;
  float* exbuf   = p; p += (size_t)ET * HEADS;

  const int mtiles  = N >> 5;                 // 625 row tiles (N % 32 == 0)
  const int gtiles  = mtiles * (HIDDEN >> 6); // 2500 wave-tiles per layer GEMM
  const int ftiles  = mtiles * (CHAN >> 6);   // 625 for final GEMM
  const int nHC     = N * HIDDEN;
  const dim3 blkG(128);                       // 4 waves/block
  const dim3 blkE(256);

  for (int b = 0; b < B; ++b) {
    const float* xb = x + (size_t)b * N * FEAT;
    float* outb = (float*)d_out + (size_t)b * N * CHAN;

    // ---- GAT layer 1: h1 = x@W1, scores, softmax over incoming edges, agg ----
    wmma_gemm_kernel<true, false><<<(gtiles + 3) / 4, blkG, 0, stream>>>(
        xb, W1, nullptr, asrc1, adst1, buf_h, ssrc, sdst, N, FEAT, HIDDEN);
    layer_init_kernel<<<(nHC + 255) / 256, blkE, 0, stream>>>(buf_agg, b1, denom, mord, N);
    edge_max_kernel<<<(ET + 255) / 256, blkE, 0, stream>>>(ei, E, N, ssrc, sdst, mord);
    edge_exp_kernel<<<(ET + 255) / 256, blkE, 0, stream>>>(ei, E, N, ssrc, sdst, mord, exbuf, denom);
    edge_agg_kernel<<<(ET + 7) / 8, blkE, 0, stream>>>(ei, E, N, exbuf, denom, buf_h, buf_agg);
    elu_kernel<<<(nHC + 255) / 256, blkE, 0, stream>>>(buf_agg, nHC);

    // ---- GAT layer 2 ----
    wmma_gemm_kernel<true, false><<<(gtiles + 3) / 4, blkG, 0, stream>>>(
        buf_agg, W2, nullptr, asrc2, adst2, buf_h, ssrc, sdst, N, HIDDEN, HIDDEN);
    layer_init_kernel<<<(nHC + 255) / 256, blkE, 0, stream>>>(buf_agg, b2, denom, mord, N);
    edge_max_kernel<<<(ET + 255) / 256, blkE, 0, stream>>>(ei, E, N, ssrc, sdst, mord);
    edge_exp_kernel<<<(ET + 255) / 256, blkE, 0, stream>>>(ei, E, N, ssrc, sdst, mord, exbuf, denom);
    edge_agg_kernel<<<(ET + 7) / 8, blkE, 0, stream>>>(ei, E, N, exbuf, denom, buf_h, buf_agg);

    // ---- final linear: out = h2 @ Wf + bf ----
    wmma_gemm_kernel<false, true><<<(ftiles + 3) / 4, blkG, 0, stream>>>(
        buf_agg, Wf, bf, nullptr, nullptr, outb, nullptr, nullptr, N, HIDDEN, CHAN);
  }
}